// OGNN_layer_16630113370191
// MI455X (gfx1250) — compile-verified
//
#include <hip/hip_runtime.h>
#include <math.h>

// ---------------------------------------------------------------------------
// OGNN layer for MI455X (gfx1250, wave32):
//   support = x @ hamilton(weight)        -> fp32 WMMA 16x16x4, 4 col-tiles/wave
//   agg[dst] += w_e * support[src]        -> global_atomic_add_f32 scatter
//   out = tanh(BN(agg) * gamma + beta)    -> streaming passes
// ---------------------------------------------------------------------------

typedef __attribute__((ext_vector_type(2))) float v2f;
typedef __attribute__((ext_vector_type(8))) float v8f;

#define N_NODES 100000
#define D 256
#define BN_EPS 1e-5f

// Octonion multiplication table, stored [c][r] (row-major as in the reference)
__constant__ int c_IDX[64] = {
    0,1,2,3,4,5,6,7,
    1,0,3,2,5,4,7,6,
    2,3,0,1,6,7,4,5,
    3,2,1,0,7,6,5,4,
    4,5,6,7,0,1,2,3,
    5,4,7,6,1,0,3,2,
    6,7,4,5,2,3,0,1,
    7,6,5,4,3,2,1,0};
__constant__ float c_SIGN[64] = {
     1, 1, 1, 1, 1, 1, 1, 1,
     1,-1, 1,-1, 1,-1,-1, 1,
     1,-1,-1, 1, 1, 1,-1,-1,
     1, 1,-1,-1, 1,-1, 1,-1,
     1,-1,-1,-1,-1, 1, 1, 1,
     1, 1,-1, 1,-1,-1,-1, 1,
     1, 1, 1,-1,-1, 1,-1,-1,
     1,-1, 1, 1,-1,-1, 1,-1};

// H[r*32+i][c*32+k] = SIGN[c][r] * weight[i][IDX[c][r]*32 + k]
__global__ void build_hamilton_kernel(const float* __restrict__ w,
                                      float* __restrict__ H) {
    int idx = blockIdx.x * 256 + threadIdx.x;   // 65536 elements
    int row = idx >> 8;
    int col = idx & 255;
    int r = row >> 5, i = row & 31;
    int c = col >> 5, k = col & 31;
    H[idx] = c_SIGN[c * 8 + r] * w[i * 256 + c_IDX[c * 8 + r] * 32 + k];
}

// support = X[100000,256] @ H[256,256], fp32 WMMA 16x16x4.
// grid = (N/32), block = 256 (8 waves). Waves 0-3 own row stripe 0 of the
// block (16 rows), waves 4-7 own row stripe 1. Each wave computes 4 adjacent
// 16x16 column tiles (64 cols), so one A fragment feeds 4 WMMAs per K-step.
//
// Per-lane fragment layout for V_WMMA_F32_16X16X4_F32 (ISA 7.12.2):
//   A 16x4:  lane<16 -> row=lane,    K={0,1}; lane>=16 -> row=lane-16, K={2,3}
//   B 4x16:  lane<16 -> col=lane,    K={0,1}; lane>=16 -> col=lane-16, K={2,3}
//   C/D:     VGPR g -> row g (lanes 0-15) / row g+8 (lanes 16-31), col=lane%16
__global__ void gemm_wmma_kernel(const float* __restrict__ X,
                                 const float* __restrict__ H,
                                 float* __restrict__ S) {
    const int lane = threadIdx.x & 31;
    const int wave = threadIdx.x >> 5;
    const int row0 = blockIdx.x * 32 + (wave >> 2) * 16;
    const int colg = (wave & 3) * 64;     // first of 4 col tiles for this wave
    const int m    = row0 + (lane & 15);
    const int nloc = lane & 15;
    const int koff = (lane >> 4) * 2;

    const float* __restrict__ xrow = X + (size_t)m * D;

    v8f c0 = {}, c1 = {}, c2 = {}, c3 = {};
    for (int k = 0; k < D; k += 4) {
        // A fragment: two consecutive K elements of row m (8B-aligned),
        // shared by all four column tiles this wave owns.
        v2f a = *(const v2f*)(xrow + k + koff);

        // B fragments: H[k+koff][col], H[k+koff+1][col] for 4 tiles.
        const float* __restrict__ h0 = H + (size_t)(k + koff) * D + colg + nloc;
        v2f b0, b1, b2, b3;
        b0.x = h0[0];   b0.y = h0[0 + D];
        b1.x = h0[16];  b1.y = h0[16 + D];
        b2.x = h0[32];  b2.y = h0[32 + D];
        b3.x = h0[48];  b3.y = h0[48 + D];

        c0 = __builtin_amdgcn_wmma_f32_16x16x4_f32(false, a, false, b0,
                                                   (short)0, c0, false, false);
        c1 = __builtin_amdgcn_wmma_f32_16x16x4_f32(false, a, false, b1,
                                                   (short)0, c1, false, false);
        c2 = __builtin_amdgcn_wmma_f32_16x16x4_f32(false, a, false, b2,
                                                   (short)0, c2, false, false);
        c3 = __builtin_amdgcn_wmma_f32_16x16x4_f32(false, a, false, b3,
                                                   (short)0, c3, false, false);
    }

    const int rbase = row0 + ((lane >> 4) << 3);
    float* __restrict__ s0 = S + (size_t)rbase * D + colg + nloc;
#pragma unroll
    for (int g = 0; g < 8; ++g) {
        s0[(size_t)g * D + 0]  = c0[g];
        s0[(size_t)g * D + 16] = c1[g];
        s0[(size_t)g * D + 32] = c2[g];
        s0[(size_t)g * D + 48] = c3[g];
    }
}

// One edge per wave32: lane handles 8 columns (two float4 loads, 8 atomics).
__global__ void aggregate_kernel(const float* __restrict__ S,
                                 const float* __restrict__ ew,
                                 const int* __restrict__ esrc,
                                 const int* __restrict__ edst,
                                 float* __restrict__ out, int E) {
    const int lane = threadIdx.x & 31;
    const int wave = threadIdx.x >> 5;
    const long long e = (long long)blockIdx.x * 8 + wave;
    if (e >= E) return;

    const int src = esrc[e];
    const int dst = edst[e];
    const float w = ew[e];

    const float4* sp = (const float4*)(S + (size_t)src * D) + lane * 2;
    float4 v0 = sp[0];
    float4 v1 = sp[1];

    float* op = out + (size_t)dst * D + lane * 8;
    atomicAdd(op + 0, w * v0.x);
    atomicAdd(op + 1, w * v0.y);
    atomicAdd(op + 2, w * v0.z);
    atomicAdd(op + 3, w * v0.w);
    atomicAdd(op + 4, w * v1.x);
    atomicAdd(op + 5, w * v1.y);
    atomicAdd(op + 6, w * v1.z);
    atomicAdd(op + 7, w * v1.w);
}

// Per-column sum / sumsq partials; thread t owns column t (coalesced rows).
__global__ void bn_stats_kernel(const float* __restrict__ A,
                                float* __restrict__ sums,
                                float* __restrict__ sqs,
                                int rows_per_block) {
    const int col = threadIdx.x;
    const int r0  = blockIdx.x * rows_per_block;
    float s = 0.f, q = 0.f;
    for (int r = 0; r < rows_per_block; ++r) {
        int row = r0 + r;
        if (row < N_NODES) {
            float v = A[(size_t)row * D + col];
            s += v;
            q += v * v;
        }
    }
    atomicAdd(&sums[col], s);
    atomicAdd(&sqs[col], q);
}

// In-place: out = tanh((agg - mean) * rsqrt(var + eps) * gamma + beta)
__global__ void bn_tanh_kernel(float* __restrict__ out,
                               const float* __restrict__ sums,
                               const float* __restrict__ sqs,
                               const float* __restrict__ gamma,
                               const float* __restrict__ beta) {
    const size_t i = (size_t)blockIdx.x * 256 + threadIdx.x;
    const int col = (int)(i & (D - 1));
    const float inv_n = 1.0f / (float)N_NODES;
    const float mean  = sums[col] * inv_n;
    const float var   = sqs[col] * inv_n - mean * mean;
    const float v     = out[i];
    const float y = (v - mean) * rsqrtf(var + BN_EPS) * gamma[col] + beta[col];
    out[i] = tanhf(y);
}

extern "C" void kernel_launch(void* const* d_in, const int* in_sizes, int n_in,
                              void* d_out, int out_size, void* d_ws, size_t ws_size,
                              hipStream_t stream) {
    const float* x      = (const float*)d_in[0];   // [100000, 256]
    const float* weight = (const float*)d_in[1];   // [32, 256]
    const float* gamma  = (const float*)d_in[2];   // [256]
    const float* beta   = (const float*)d_in[3];   // [256]
    const float* ew     = (const float*)d_in[4];   // [E]
    const int*   esrc   = (const int*)d_in[5];     // [E]
    const int*   edst   = (const int*)d_in[6];     // [E]
    const int E = in_sizes[4];

    // Workspace layout: H [256*256] | support [100000*256] | sums[256] | sqs[256]
    float* H       = (float*)d_ws;
    float* support = H + 256 * 256;
    float* sums    = support + (size_t)N_NODES * D;
    float* sqs     = sums + 256;

    float* out = (float*)d_out;

    // Zero the atomic targets (harness poisons them with 0xAA).
    hipMemsetAsync(out, 0, (size_t)N_NODES * D * sizeof(float), stream);
    hipMemsetAsync(sums, 0, 2 * 256 * sizeof(float), stream);

    // 1) Hamilton matrix (256x256)
    build_hamilton_kernel<<<256, 256, 0, stream>>>(weight, H);

    // 2) support = x @ H via fp32 WMMA 16x16x4 (4 column tiles per wave)
    gemm_wmma_kernel<<<N_NODES / 32, 256, 0, stream>>>(x, H, support);

    // 3) edge scatter: out[dst] += w * support[src]
    aggregate_kernel<<<(E + 7) / 8, 256, 0, stream>>>(support, ew, esrc, edst, out, E);

    // 4) BN stats (400 blocks x 250 rows)
    const int rows_per_block = 250;
    bn_stats_kernel<<<(N_NODES + rows_per_block - 1) / rows_per_block, 256, 0, stream>>>(
        out, sums, sqs, rows_per_block);

    // 5) normalize + affine + tanh (in place on d_out)
    bn_tanh_kernel<<<(N_NODES * D) / 256, 256, 0, stream>>>(out, sums, sqs, gamma, beta);
}